// MyLightGCN_28475633172847
// MI455X (gfx1250) — compile-verified
//
#include <hip/hip_runtime.h>
#include <math.h>

#define N_NODES 100000
#define N_TOTAL 150000
#define D_IN    512
#define H_LAT   256
#define N_CLASS 40
#define NNZ_E   4800000
#define NUM_LAYERS 3
#define SCAN_B  1024

typedef float v2f __attribute__((ext_vector_type(2)));
typedef float v4f __attribute__((ext_vector_type(4)));
typedef float v8f __attribute__((ext_vector_type(8)));

// ---------------------------------------------------------------------------
// Kernel 1: emb = X @ W_red + b_red   (fp32 WMMA 16x16x4)
// One block = 16 rows of X staged in LDS; 8 waves, each owns 2 N-tiles.
// Writes result into both acc and cur.
// ---------------------------------------------------------------------------
__global__ __launch_bounds__(256) void hg_gemm_emb(
    const float* __restrict__ X, const float* __restrict__ W,
    const float* __restrict__ bias, float* __restrict__ acc,
    float* __restrict__ cur) {
  __shared__ float lds[16 * D_IN];  // 32 KB
  const int tid = threadIdx.x;
  const size_t row_base = (size_t)blockIdx.x * 16;

  const v4f* __restrict__ Xv = (const v4f*)(X + row_base * D_IN);
  v4f* Lv = (v4f*)lds;
#pragma unroll
  for (int i = 0; i < 8; ++i) Lv[tid + i * 256] = Xv[tid + i * 256];
  __syncthreads();

  const int wave = tid >> 5;
  const int lane = tid & 31;
  const int m  = lane & 15;
  const int kb = (lane >> 4) << 1;
  const int n  = (wave << 5) + m;

  v8f c0 = {0.f, 0.f, 0.f, 0.f, 0.f, 0.f, 0.f, 0.f};
  v8f c1 = {0.f, 0.f, 0.f, 0.f, 0.f, 0.f, 0.f, 0.f};

  for (int k = 0; k < D_IN; k += 4) {
    v2f a = *(const v2f*)&lds[m * D_IN + k + kb];
    const float* __restrict__ W0 = W + (size_t)(k + kb) * H_LAT;
    v2f b0, b1;
    b0.x = W0[n];          b0.y = W0[H_LAT + n];
    b1.x = W0[n + 16];     b1.y = W0[H_LAT + n + 16];
    c0 = __builtin_amdgcn_wmma_f32_16x16x4_f32(false, a, false, b0,
                                               (short)0, c0, false, false);
    c1 = __builtin_amdgcn_wmma_f32_16x16x4_f32(false, a, false, b1,
                                               (short)0, c1, false, false);
  }

  const int mo = (lane >> 4) << 3;
  const float bb0 = bias[n];
  const float bb1 = bias[n + 16];
#pragma unroll
  for (int r = 0; r < 8; ++r) {
    const size_t row = row_base + (size_t)(r + mo);
    const float e0 = c0[r] + bb0;
    const float e1 = c1[r] + bb1;
    acc[row * H_LAT + n]      = e0;
    cur[row * H_LAT + n]      = e0;
    acc[row * H_LAT + n + 16] = e1;
    cur[row * H_LAT + n + 16] = e1;
  }
}

// ---------------------------------------------------------------------------
// CSR construction (done once, reused for all 3 SPMM layers)
// ---------------------------------------------------------------------------
__global__ __launch_bounds__(256) void hg_zero_u32(unsigned* __restrict__ p,
                                                   int n) {
  const int i = blockIdx.x * 256 + threadIdx.x;
  if (i < n) p[i] = 0u;
}

__global__ __launch_bounds__(256) void hg_hist(const int* __restrict__ rows,
                                               unsigned* __restrict__ cnt) {
  const int e = blockIdx.x * 256 + threadIdx.x;
  if (e < NNZ_E) atomicAdd(&cnt[rows[e]], 1u);
}

// Per-block exclusive scan (Hillis-Steele over 1024 elems) + block totals.
__global__ __launch_bounds__(SCAN_B) void hg_scan_block(
    const unsigned* __restrict__ cnt, unsigned* __restrict__ excl,
    unsigned* __restrict__ bsum, int n) {
  __shared__ unsigned s[SCAN_B];
  const int tid = threadIdx.x;
  const int i = blockIdx.x * SCAN_B + tid;
  const unsigned v = (i < n) ? cnt[i] : 0u;
  s[tid] = v;
  __syncthreads();
  for (int off = 1; off < SCAN_B; off <<= 1) {
    const unsigned t = (tid >= off) ? s[tid - off] : 0u;
    __syncthreads();
    s[tid] += t;
    __syncthreads();
  }
  if (i < n) excl[i] = s[tid] - v;            // exclusive
  if (tid == SCAN_B - 1) bsum[blockIdx.x] = s[SCAN_B - 1];
}

// Single-block exclusive scan of the block totals (nb <= 1024).
__global__ __launch_bounds__(SCAN_B) void hg_scan_sums(
    unsigned* __restrict__ bsum, int nb) {
  __shared__ unsigned s[SCAN_B];
  const int tid = threadIdx.x;
  const unsigned v = (tid < nb) ? bsum[tid] : 0u;
  s[tid] = v;
  __syncthreads();
  for (int off = 1; off < SCAN_B; off <<= 1) {
    const unsigned t = (tid >= off) ? s[tid - off] : 0u;
    __syncthreads();
    s[tid] += t;
    __syncthreads();
  }
  if (tid < nb) bsum[tid] = s[tid] - v;       // exclusive
}

__global__ __launch_bounds__(256) void hg_scan_add(
    unsigned* __restrict__ excl, const unsigned* __restrict__ bsum, int n) {
  const int i = blockIdx.x * 256 + threadIdx.x;
  if (i < n) excl[i] += bsum[i / SCAN_B];
}

__global__ __launch_bounds__(256) void hg_scatter(
    const float* __restrict__ vals, const int* __restrict__ rows,
    const int* __restrict__ cols, const unsigned* __restrict__ rptr,
    unsigned* __restrict__ fill, int* __restrict__ scol,
    float* __restrict__ sval) {
  const int e = blockIdx.x * 256 + threadIdx.x;
  if (e >= NNZ_E) return;
  const int r = rows[e];
  const unsigned pos = rptr[r] + atomicAdd(&fill[r], 1u);
  scol[pos] = cols[e];
  sval[pos] = vals[e];
}

// ---------------------------------------------------------------------------
// Gather-form SPMM: one wave per row, 256 features in 8 VGPRs/lane.
// Fuses nxt = A @ cur  and  acc += nxt. No atomics, no zeroing pass.
// Gathers hit the L2-resident (154 MB < 192 MB) embedding table.
// ---------------------------------------------------------------------------
__global__ __launch_bounds__(256) void hg_spmm_csr(
    const unsigned* __restrict__ rptr, const int* __restrict__ scol,
    const float* __restrict__ sval, const float* __restrict__ cur,
    float* __restrict__ nxt, float* __restrict__ acc) {
  const size_t w = ((size_t)blockIdx.x * 256 + threadIdx.x) >> 5;
  const int lane = threadIdx.x & 31;
  if (w >= (size_t)N_TOTAL) return;
  const unsigned i0 = rptr[w];
  const unsigned i1 = (w == (size_t)(N_TOTAL - 1)) ? (unsigned)NNZ_E
                                                   : rptr[w + 1];
  float f[8] = {0.f, 0.f, 0.f, 0.f, 0.f, 0.f, 0.f, 0.f};
  for (unsigned i = i0; i < i1; ++i) {
    const int c = scol[i];
    const float v = sval[i];
    const float* __restrict__ src = cur + (size_t)c * H_LAT + lane;
#pragma unroll
    for (int j = 0; j < 8; ++j) f[j] = fmaf(v, src[32 * j], f[j]);
  }
  float* __restrict__ nd = nxt + w * H_LAT + lane;
  float* __restrict__ ad = acc + w * H_LAT + lane;
#pragma unroll
  for (int j = 0; j < 8; ++j) {
    nd[32 * j] = f[j];
    ad[32 * j] += f[j];
  }
}

// ---------------------------------------------------------------------------
// Classifier: Z = (acc[:N_NODES] * 0.25) @ W_cls + b_cls  (fp32 WMMA, N->48)
// ---------------------------------------------------------------------------
__global__ __launch_bounds__(256) void hg_cls(
    const float* __restrict__ acc, const float* __restrict__ Wc,
    const float* __restrict__ bc, float* __restrict__ Z) {
  const int wave = threadIdx.x >> 5;
  const int lane = threadIdx.x & 31;
  const size_t tile = (size_t)blockIdx.x * 8 + wave;
  if (tile >= (size_t)(N_NODES / 16)) return;   // wave-uniform
  const size_t row_base = tile * 16;

  const int m  = lane & 15;
  const int kb = (lane >> 4) << 1;
  const int n  = lane & 15;

  v8f c0 = {0.f, 0.f, 0.f, 0.f, 0.f, 0.f, 0.f, 0.f};
  v8f c1 = c0, c2 = c0;
  const float* __restrict__ Arow = acc + (row_base + (size_t)m) * H_LAT;
  const bool tail = (n + 32) < N_CLASS;

  for (int k = 0; k < H_LAT; k += 4) {
    v2f a = *(const v2f*)&Arow[k + kb];
    a.x *= 0.25f; a.y *= 0.25f;
    const float* __restrict__ W0 = Wc + (size_t)(k + kb) * N_CLASS;
    const float* __restrict__ W1 = W0 + N_CLASS;
    v2f b0, b1, b2;
    b0.x = W0[n];       b0.y = W1[n];
    b1.x = W0[n + 16];  b1.y = W1[n + 16];
    b2.x = tail ? W0[n + 32] : 0.f;
    b2.y = tail ? W1[n + 32] : 0.f;
    c0 = __builtin_amdgcn_wmma_f32_16x16x4_f32(false, a, false, b0,
                                               (short)0, c0, false, false);
    c1 = __builtin_amdgcn_wmma_f32_16x16x4_f32(false, a, false, b1,
                                               (short)0, c1, false, false);
    c2 = __builtin_amdgcn_wmma_f32_16x16x4_f32(false, a, false, b2,
                                               (short)0, c2, false, false);
  }

  const int mo = (lane >> 4) << 3;
  const float bc0 = bc[n];
  const float bc1 = bc[n + 16];
  const float bc2 = tail ? bc[n + 32] : 0.f;
#pragma unroll
  for (int r = 0; r < 8; ++r) {
    const size_t row = row_base + (size_t)(r + mo);
    Z[row * N_CLASS + n]      = c0[r] + bc0;
    Z[row * N_CLASS + n + 16] = c1[r] + bc1;
    if (tail) Z[row * N_CLASS + n + 32] = c2[r] + bc2;
  }
}

// ---------------------------------------------------------------------------
// In-place log_softmax over 40 classes, one thread per row.
// ---------------------------------------------------------------------------
__global__ __launch_bounds__(256) void hg_lsm(float* __restrict__ Z) {
  const size_t row = (size_t)blockIdx.x * 256 + threadIdx.x;
  if (row >= (size_t)N_NODES) return;
  float* __restrict__ z = Z + row * N_CLASS;
  float zr[N_CLASS];
  float mx = -INFINITY;
#pragma unroll
  for (int i = 0; i < N_CLASS; ++i) { zr[i] = z[i]; mx = fmaxf(mx, zr[i]); }
  float s = 0.f;
#pragma unroll
  for (int i = 0; i < N_CLASS; ++i) s += __expf(zr[i] - mx);
  const float lse = mx + __logf(s);
#pragma unroll
  for (int i = 0; i < N_CLASS; ++i) z[i] = zr[i] - lse;
}

// ---------------------------------------------------------------------------
extern "C" void kernel_launch(void* const* d_in, const int* in_sizes, int n_in,
                              void* d_out, int out_size, void* d_ws,
                              size_t ws_size, hipStream_t stream) {
  const float* X     = (const float*)d_in[0];
  const float* W_red = (const float*)d_in[1];
  const float* b_red = (const float*)d_in[2];
  const float* W_cls = (const float*)d_in[3];
  const float* b_cls = (const float*)d_in[4];
  const float* evals = (const float*)d_in[5];
  const int*   erows = (const int*)d_in[6];
  const int*   ecols = (const int*)d_in[7];
  float* out = (float*)d_out;

  const size_t NE = (size_t)N_TOTAL * H_LAT;     // 38.4M floats / buffer
  float*    acc   = (float*)d_ws;
  float*    bufA  = acc + NE;
  float*    bufB  = bufA + NE;
  float*    sval  = bufB + NE;                    // 4.8M
  int*      scol  = (int*)(sval + NNZ_E);         // 4.8M
  unsigned* cnt   = (unsigned*)(scol + NNZ_E);    // 150000
  unsigned* rptr  = cnt + N_TOTAL;                // 150000
  unsigned* fill  = rptr + N_TOTAL;               // 150000
  unsigned* bsums = fill + N_TOTAL;               // <=1024

  const int nscan  = (N_TOTAL + SCAN_B - 1) / SCAN_B;       // 147
  const int nblk_r = (N_TOTAL + 255) / 256;                 // 586
  const int nblk_e = NNZ_E / 256;                           // 18750

  // ---- build CSR once (reused by all layers) ----
  hg_zero_u32<<<nblk_r, 256, 0, stream>>>(cnt, N_TOTAL);
  hg_zero_u32<<<nblk_r, 256, 0, stream>>>(fill, N_TOTAL);
  hg_hist<<<nblk_e, 256, 0, stream>>>(erows, cnt);
  hg_scan_block<<<nscan, SCAN_B, 0, stream>>>(cnt, rptr, bsums, N_TOTAL);
  hg_scan_sums<<<1, SCAN_B, 0, stream>>>(bsums, nscan);
  hg_scan_add<<<nblk_r, 256, 0, stream>>>(rptr, bsums, N_TOTAL);
  hg_scatter<<<nblk_e, 256, 0, stream>>>(evals, erows, ecols, rptr, fill,
                                         scol, sval);

  // ---- emb -> acc and cur (fp32 WMMA GEMM) ----
  hg_gemm_emb<<<N_TOTAL / 16, 256, 0, stream>>>(X, W_red, b_red, acc, bufA);

  // ---- 3 propagation layers: fused gather-SPMM + acc update ----
  float* cur = bufA;
  float* nxt = bufB;
  const int nblk_s = (N_TOTAL * 32 + 255) / 256;            // 18750
  for (int l = 0; l < NUM_LAYERS; ++l) {
    hg_spmm_csr<<<nblk_s, 256, 0, stream>>>(rptr, scol, sval, cur, nxt, acc);
    float* t = cur; cur = nxt; nxt = t;
  }

  // ---- classifier (fused /4, +b_cls) into d_out, then in-place LSM ----
  hg_cls<<<(N_NODES / 16 + 7) / 8, 256, 0, stream>>>(acc, W_cls, b_cls, out);
  hg_lsm<<<(N_NODES + 255) / 256, 256, 0, stream>>>(out);
}